// BackboneWithEmbedding_53618371724117
// MI455X (gfx1250) — compile-verified
//
#include <hip/hip_runtime.h>
#include <math.h>

#define P_N   4096
#define FIN   64
#define HG_N  64
#define C_N   64
#define T_N   32
#define E_N   65536
#define HOUT_N 12
#define LEAKY 0.2f
#define LN_EPS 1e-5f

typedef float v2f __attribute__((ext_vector_type(2)));
typedef float v8f __attribute__((ext_vector_type(8)));

// ---------------- workspace layout (floats) ----------------
// HL  : T*P*HG
// HR  : T*P*HG   (reused as Y after edge passes)
// NUM : T*P*HG   (becomes X_hat)
// EV  : T*E
// EMAX: T*P      (monotone-uint keys)
// DEN : T*P
static const size_t OFF_HL   = 0;
static const size_t OFF_HR   = OFF_HL  + (size_t)T_N * P_N * HG_N;
static const size_t OFF_NUM  = OFF_HR  + (size_t)T_N * P_N * HG_N;
static const size_t OFF_EV   = OFF_NUM + (size_t)T_N * P_N * HG_N;
static const size_t OFF_EMAX = OFF_EV  + (size_t)T_N * E_N;
static const size_t OFF_DEN  = OFF_EMAX + (size_t)T_N * P_N;
static const size_t WS_TOTAL = OFF_DEN + (size_t)T_N * P_N;

__global__ void k_zero(float* __restrict__ p, size_t n) {
    size_t i = (size_t)blockIdx.x * blockDim.x + threadIdx.x;
    if (i < n) p[i] = 0.0f;
}

// ---- Stage 1: HL = x@Wl, HR = x@Wr for all t, via V_WMMA_F32_16X16X4_F32 ----
// X layout: X[((p*FIN)+f)*T + t].  One wave computes a 16-row x 64-col tile of both.
// Wl/Wr are staged TRANSPOSED in LDS so each B fragment is one aligned 8B ds load.
__global__ __launch_bounds__(256) void k_gemm_hlr(
    const float* __restrict__ X, const float* __restrict__ Wl,
    const float* __restrict__ Wr, float* __restrict__ HL, float* __restrict__ HR)
{
    __shared__ float sWlT[FIN * HG_N];   // sWlT[n*64 + k] = Wl[k*64 + n]
    __shared__ float sWrT[FIN * HG_N];

    int tid = threadIdx.x;
#pragma unroll
    for (int j = 0; j < 16; ++j) {
        int e = tid + (j << 8);          // 0..4095, coalesced global reads
        int k = e >> 6, n = e & 63;
        sWlT[(n << 6) + k] = Wl[e];
        sWrT[(n << 6) + k] = Wr[e];
    }
    __syncthreads();

    int wave = (blockIdx.x * (int)blockDim.x + (int)threadIdx.x) >> 5;
    int lane = threadIdx.x & 31;
    int t  = wave >> 8;              // 256 row-tiles per timestep
    int p0 = (wave & 255) << 4;
    int m     = lane & 15;           // A row within tile
    int khalf = (lane >> 4) << 1;    // 0 or 2 (K sub-offset per ISA layout)
    int nlane = lane & 15;

    v8f accL[4], accR[4];
#pragma unroll
    for (int j = 0; j < 4; ++j)
#pragma unroll
        for (int r = 0; r < 8; ++r) { accL[j][r] = 0.0f; accR[j][r] = 0.0f; }

#pragma unroll 4
    for (int k0 = 0; k0 < FIN; k0 += 4) {
        int kk = k0 + khalf;
        const float* xp = X + ((size_t)(p0 + m) * FIN + kk) * T_N + t;
        v2f a; a.x = xp[0]; a.y = xp[T_N];   // f and f+1 (stride T)
#pragma unroll
        for (int j = 0; j < 4; ++j) {
            int n = (j << 4) + nlane;
            v2f bl = *(const v2f*)&sWlT[(n << 6) + kk];   // ds_load_b64
            accL[j] = __builtin_amdgcn_wmma_f32_16x16x4_f32(
                false, a, false, bl, (short)0, accL[j], false, false);
            v2f br = *(const v2f*)&sWrT[(n << 6) + kk];
            accR[j] = __builtin_amdgcn_wmma_f32_16x16x4_f32(
                false, a, false, br, (short)0, accR[j], false, false);
        }
    }
    int mbase = (lane >> 4) << 3;    // D rows 0-7 (lanes 0-15) or 8-15
#pragma unroll
    for (int j = 0; j < 4; ++j)
#pragma unroll
        for (int r = 0; r < 8; ++r) {
            size_t idx = ((size_t)t * P_N + p0 + mbase + r) * HG_N + (j << 4) + nlane;
            HL[idx] = accL[j][r];
            HR[idx] = accR[j][r];
        }
}

// ---- Stage 2: per-edge attention logit + segment max (wave per edge) ----
__device__ __forceinline__ unsigned f32_to_key(float f) {
    unsigned fb = __float_as_uint(f);
    return (fb & 0x80000000u) ? ~fb : (fb | 0x80000000u);
}
__device__ __forceinline__ float key_to_f32(unsigned k) {
    unsigned fb = (k & 0x80000000u) ? (k ^ 0x80000000u) : ~k;
    return __uint_as_float(fb);
}

__global__ __launch_bounds__(256) void k_edge_max(
    const float* __restrict__ HL, const float* __restrict__ HR,
    const int* __restrict__ EIDX, const float* __restrict__ att,
    float* __restrict__ EV, unsigned* __restrict__ EMAXK)
{
    size_t wave = (((size_t)blockIdx.x * blockDim.x) + threadIdx.x) >> 5;
    int lane = threadIdx.x & 31;
    int t = (int)(wave >> 16);            // E_N = 65536 edges per t
    int e = (int)(wave & (E_N - 1));
    const int* ei = EIDX + (size_t)t * 2 * E_N;
    int src = ei[e];
    int dst = ei[E_N + e];
    int c = lane << 1;                    // 2 channels per lane (wave32, HG=64)
    const float* hlp = HL + ((size_t)t * P_N + src) * HG_N + c;
    const float* hrp = HR + ((size_t)t * P_N + dst) * HG_N + c;
    float s = 0.0f;
#pragma unroll
    for (int u = 0; u < 2; ++u) {
        float z = hlp[u] + hrp[u];
        z = z > 0.0f ? z : LEAKY * z;
        s += z * att[c + u];
    }
#pragma unroll
    for (int off = 16; off > 0; off >>= 1) s += __shfl_xor(s, off, 32);
    if (lane == 0) {
        EV[(size_t)t * E_N + e] = s;
        atomicMax(&EMAXK[(size_t)t * P_N + dst], f32_to_key(s));
    }
}

// ---- Stage 3: exp / segment-sum accumulation (wave per edge) ----
__global__ __launch_bounds__(256) void k_edge_accum(
    const float* __restrict__ HL, const int* __restrict__ EIDX,
    const float* __restrict__ EV, const unsigned* __restrict__ EMAXK,
    float* __restrict__ DEN, float* __restrict__ NUM)
{
    size_t wave = (((size_t)blockIdx.x * blockDim.x) + threadIdx.x) >> 5;
    int lane = threadIdx.x & 31;
    int t = (int)(wave >> 16);
    int e = (int)(wave & (E_N - 1));
    const int* ei = EIDX + (size_t)t * 2 * E_N;
    int src = ei[e];
    int dst = ei[E_N + e];
    float mx = key_to_f32(EMAXK[(size_t)t * P_N + dst]);
    float ex = __expf(EV[(size_t)t * E_N + e] - mx);
    if (lane == 0) atomicAdd(&DEN[(size_t)t * P_N + dst], ex);
    int c = lane << 1;
    const float* hlp = HL + ((size_t)t * P_N + src) * HG_N + c;
    float* np = NUM + ((size_t)t * P_N + dst) * HG_N + c;
    atomicAdd(&np[0], ex * hlp[0]);
    atomicAdd(&np[1], ex * hlp[1]);
}

// ---- Stage 4: X_hat = relu(num/den + gat_b), in place in NUM ----
__global__ void k_finalize_xhat(float* __restrict__ NUM, const float* __restrict__ DEN,
                                const float* __restrict__ gat_b)
{
    size_t i = (size_t)blockIdx.x * blockDim.x + threadIdx.x;
    if (i >= (size_t)T_N * P_N * HG_N) return;
    int c = (int)(i & (HG_N - 1));
    float d = DEN[i >> 6];
    float v = (d > 0.0f) ? NUM[i] / d : 0.0f;
    v += gat_b[c];
    NUM[i] = v > 0.0f ? v : 0.0f;
}

// ---- Stage 5: Y = relu(Xhat@W_time^T + x@W_res^T + b_time + b_res), WMMA K=128 ----
// W_time/W_res staged in LDS in natural layout: fragment pair W[n*64+kk..kk+1] is contiguous.
__global__ __launch_bounds__(256) void k_gemm_fused(
    const float* __restrict__ XHAT, const float* __restrict__ Xin,
    const float* __restrict__ W_time, const float* __restrict__ b_time,
    const float* __restrict__ W_res, const float* __restrict__ b_res,
    float* __restrict__ Y)
{
    __shared__ float sWt[C_N * HG_N];
    __shared__ float sWr[C_N * FIN];

    int tid = threadIdx.x;
#pragma unroll
    for (int j = 0; j < 16; ++j) {
        int e = tid + (j << 8);
        sWt[e] = W_time[e];
        sWr[e] = W_res[e];
    }
    __syncthreads();

    int wave = (blockIdx.x * (int)blockDim.x + (int)threadIdx.x) >> 5;
    int lane = threadIdx.x & 31;
    int t  = wave >> 8;
    int p0 = (wave & 255) << 4;
    int m     = lane & 15;
    int khalf = (lane >> 4) << 1;
    int nlane = lane & 15;

    v8f acc[4];
#pragma unroll
    for (int j = 0; j < 4; ++j)
#pragma unroll
        for (int r = 0; r < 8; ++r) acc[j][r] = 0.0f;

    // time-conv half: A = X_hat (row-contig), B[k][n] = W_time[n][k]
#pragma unroll 4
    for (int k0 = 0; k0 < HG_N; k0 += 4) {
        int kk = k0 + khalf;
        const float* ap = XHAT + ((size_t)t * P_N + p0 + m) * HG_N + kk;
        v2f a; a.x = ap[0]; a.y = ap[1];
#pragma unroll
        for (int j = 0; j < 4; ++j) {
            int n = (j << 4) + nlane;
            v2f b = *(const v2f*)&sWt[(n << 6) + kk];      // ds_load_b64
            acc[j] = __builtin_amdgcn_wmma_f32_16x16x4_f32(
                false, a, false, b, (short)0, acc[j], false, false);
        }
    }
    // residual half: A = x (stride-T), B[k][n] = W_res[n][k]
#pragma unroll 4
    for (int k0 = 0; k0 < FIN; k0 += 4) {
        int kk = k0 + khalf;
        const float* ap = Xin + ((size_t)(p0 + m) * FIN + kk) * T_N + t;
        v2f a; a.x = ap[0]; a.y = ap[T_N];
#pragma unroll
        for (int j = 0; j < 4; ++j) {
            int n = (j << 4) + nlane;
            v2f b = *(const v2f*)&sWr[(n << 6) + kk];
            acc[j] = __builtin_amdgcn_wmma_f32_16x16x4_f32(
                false, a, false, b, (short)0, acc[j], false, false);
        }
    }
    int mbase = (lane >> 4) << 3;
#pragma unroll
    for (int j = 0; j < 4; ++j) {
        int col = (j << 4) + nlane;
        float bias = b_time[col] + b_res[col];
#pragma unroll
        for (int r = 0; r < 8; ++r) {
            float v = acc[j][r] + bias;
            Y[((size_t)t * P_N + p0 + mbase + r) * C_N + col] = v > 0.0f ? v : 0.0f;
        }
    }
}

// ---- Stage 6: LayerNorm over C per (t,p) row, in place ----
__global__ void k_layernorm(float* __restrict__ Y, const float* __restrict__ g,
                            const float* __restrict__ b)
{
    size_t i = (size_t)blockIdx.x * blockDim.x + threadIdx.x;
    if (i >= (size_t)T_N * P_N) return;
    float* row = Y + i * C_N;
    float mu = 0.0f;
#pragma unroll 8
    for (int c = 0; c < C_N; ++c) mu += row[c];
    mu *= (1.0f / C_N);
    float var = 0.0f;
#pragma unroll 8
    for (int c = 0; c < C_N; ++c) { float d = row[c] - mu; var += d * d; }
    var *= (1.0f / C_N);
    float inv = rsqrtf(var + LN_EPS);
#pragma unroll 8
    for (int c = 0; c < C_N; ++c) row[c] = (row[c] - mu) * inv * g[c] + b[c];
}

// ---- Stage 7: out[p][o] = sum_{t,c} yn[t][p][c] * W_final[o][t][c] + b_final[o] ----
__global__ void k_final(const float* __restrict__ Y, const float* __restrict__ Wf,
                        const float* __restrict__ bf, float* __restrict__ out)
{
    int i = blockIdx.x * (int)blockDim.x + (int)threadIdx.x;
    if (i >= P_N * HOUT_N) return;
    int o = i % HOUT_N;
    int p = i / HOUT_N;
    float s = bf[o];
    for (int t = 0; t < T_N; ++t) {
        const float* yr = Y + ((size_t)t * P_N + p) * C_N;
        const float* wr = Wf + ((size_t)o * T_N + t) * C_N;
#pragma unroll 16
        for (int c = 0; c < C_N; ++c) s += yr[c] * wr[c];
    }
    out[(size_t)p * HOUT_N + o] = s;
}

extern "C" void kernel_launch(void* const* d_in, const int* in_sizes, int n_in,
                              void* d_out, int out_size, void* d_ws, size_t ws_size,
                              hipStream_t stream) {
    const float* X       = (const float*)d_in[0];
    const int*   EIDX    = (const int*)  d_in[1];
    const float* Wl      = (const float*)d_in[2];
    const float* Wr      = (const float*)d_in[3];
    const float* att     = (const float*)d_in[4];
    const float* gat_b   = (const float*)d_in[5];
    const float* W_time  = (const float*)d_in[6];
    const float* b_time  = (const float*)d_in[7];
    const float* W_res   = (const float*)d_in[8];
    const float* b_res   = (const float*)d_in[9];
    const float* ln_g    = (const float*)d_in[10];
    const float* ln_b    = (const float*)d_in[11];
    const float* W_final = (const float*)d_in[12];
    const float* b_final = (const float*)d_in[13];
    float* out = (float*)d_out;

    float* ws   = (float*)d_ws;
    float* HL   = ws + OFF_HL;
    float* HR   = ws + OFF_HR;   // reused as Y after edge passes
    float* NUM  = ws + OFF_NUM;  // becomes X_hat
    float* EV   = ws + OFF_EV;
    unsigned* EMAXK = (unsigned*)(ws + OFF_EMAX);
    float* DEN  = ws + OFF_DEN;
    float* Y    = HR;

    // zero NUM + EV + EMAX + DEN (contiguous span)
    {
        size_t n = WS_TOTAL - OFF_NUM;
        k_zero<<<(unsigned)((n + 255) / 256), 256, 0, stream>>>(NUM, n);
    }
    // HL/HR GEMMs: T*(P/16) = 8192 waves, 8 waves/block
    k_gemm_hlr<<<1024, 256, 0, stream>>>(X, Wl, Wr, HL, HR);
    // edge passes: T*E = 2,097,152 waves, 8 waves/block
    k_edge_max<<<262144, 256, 0, stream>>>(HL, HR, EIDX, att, EV, EMAXK);
    k_edge_accum<<<262144, 256, 0, stream>>>(HL, EIDX, EV, EMAXK, DEN, NUM);
    // finalize X_hat
    k_finalize_xhat<<<(T_N * P_N * HG_N) / 256, 256, 0, stream>>>(NUM, DEN, gat_b);
    // fused time/residual convs
    k_gemm_fused<<<1024, 256, 0, stream>>>(NUM, X, W_time, b_time, W_res, b_res, Y);
    // layernorm
    k_layernorm<<<(T_N * P_N) / 256, 256, 0, stream>>>(Y, ln_g, ln_b);
    // final projection
    k_final<<<(P_N * HOUT_N) / 256, 256, 0, stream>>>(Y, W_final, b_final, out);
}